// Encoder_multi_25752623907307
// MI455X (gfx1250) — compile-verified
//
#include <hip/hip_runtime.h>

#define DF   128
#define LDIM 384   // L*D
#define NLAYERS 3

typedef float v2f __attribute__((ext_vector_type(2)));
typedef float v8f __attribute__((ext_vector_type(8)));

// ---------------- utility kernels ----------------

__global__ void zero_f(float* __restrict__ p, long n) {
  long i = (long)blockIdx.x * blockDim.x + threadIdx.x;
  long st = (long)gridDim.x * blockDim.x;
  for (; i < n; i += st) p[i] = 0.0f;
}

__global__ void count_nodes(const int* __restrict__ batch, float* __restrict__ counts, int n) {
  int i = blockIdx.x * blockDim.x + threadIdx.x;
  int st = gridDim.x * blockDim.x;
  for (; i < n; i += st) atomicAdd(&counts[batch[i]], 1.0f);
}

__global__ void degree_edges(const int* __restrict__ dst, float* __restrict__ deg, int e) {
  int i = blockIdx.x * blockDim.x + threadIdx.x;
  int st = gridDim.x * blockDim.x;
  for (; i < e; i += st) atomicAdd(&deg[dst[i]], 1.0f);
}

// deg -> dinv in place (self loop adds +1, so always > 0)
__global__ void make_dinv(float* __restrict__ d, int n) {
  int i = blockIdx.x * blockDim.x + threadIdx.x;
  int st = gridDim.x * blockDim.x;
  for (; i < n; i += st) d[i] = rsqrtf(d[i] + 1.0f);
}

// ---------------- WMMA f32 GEMM: C[n x 128] = A[n x 128] @ W[128 x 128] ----------------
// one wave per 16-row slab; 8 accumulator tiles (16x16) across the 128 columns

__global__ __launch_bounds__(128) void gemm_wmma_f32(const float* __restrict__ A,
                                                     const float* __restrict__ W,
                                                     float* __restrict__ C, int nrows) {
  const int lane = threadIdx.x & 31;
  const int wave = threadIdx.x >> 5;
  const int tile = blockIdx.x * 4 + wave;
  const int row0 = tile * 16;
  if (row0 >= nrows) return;                 // wave-uniform branch, safe for WMMA

  const int r    = lane & 15;
  const int half = lane >> 4;                // 0: K=0,1   1: K=2,3  (16x4 f32 A layout)
  int rowA = row0 + r;
  if (rowA >= nrows) rowA = nrows - 1;       // clamped load for (unused) tail rows
  const float* aptr = A + (size_t)rowA * DF;

  const v8f zero = {0.f, 0.f, 0.f, 0.f, 0.f, 0.f, 0.f, 0.f};
  v8f acc[8];
#pragma unroll
  for (int t = 0; t < 8; ++t) acc[t] = zero;

  for (int k = 0; k < DF; k += 4) {
    v2f a;
    a.x = aptr[k + 2 * half];
    a.y = aptr[k + 2 * half + 1];
    const float* w0 = W + (size_t)(k + 2 * half) * DF;
    const float* w1 = w0 + DF;
#pragma unroll
    for (int t = 0; t < 8; ++t) {
      v2f b;
      b.x = w0[t * 16 + r];                  // B[K=2h  ][col]
      b.y = w1[t * 16 + r];                  // B[K=2h+1][col]
      acc[t] = __builtin_amdgcn_wmma_f32_16x16x4_f32(
          false, a, false, b, (short)0, acc[t], false, false);
    }
  }

  // D layout: VGPR v holds M = v + 8*half, column = t*16 + r
#pragma unroll
  for (int t = 0; t < 8; ++t) {
#pragma unroll
    for (int v = 0; v < 8; ++v) {
      int row = row0 + v + 8 * half;
      if (row < nrows) C[(size_t)row * DF + t * 16 + r] = acc[t][v];
    }
  }
}

// ---------------- aggregation ----------------

// out[v][d] = hW[v][d] * dinv[v]^2 + bias[d]   (self loop + bias; also zero-inits)
__global__ void init_self_bias(const float* __restrict__ hW, const float* __restrict__ dinv,
                               const float* __restrict__ bias, float* __restrict__ out, int n) {
  long total = (long)n * DF;
  long i  = (long)blockIdx.x * blockDim.x + threadIdx.x;
  long st = (long)gridDim.x * blockDim.x;
  for (; i < total; i += st) {
    long row = i >> 7;
    int  d   = (int)(i & 127);
    float di = dinv[row];
    out[i] = hW[i] * di * di + bias[d];
  }
}

// one wave per edge: out[t][:] += hW[s][:] * dinv[s]*dinv[t]
__global__ void agg_edges(const float* __restrict__ hW, const float* __restrict__ dinv,
                          const int* __restrict__ src, const int* __restrict__ dst,
                          float* __restrict__ out, int e) {
  int lane = threadIdx.x & 31;
  int wid  = (blockIdx.x * blockDim.x + threadIdx.x) >> 5;
  int nw   = (gridDim.x * blockDim.x) >> 5;
  for (int ed = wid; ed < e; ed += nw) {
    int s = src[ed], t = dst[ed];
    float w = dinv[s] * dinv[t];
    const float4* hp = reinterpret_cast<const float4*>(hW + (size_t)s * DF);
    float4 v = hp[lane];
    float* op = out + (size_t)t * DF + lane * 4;
    atomicAdd(op + 0, v.x * w);
    atomicAdd(op + 1, v.y * w);
    atomicAdd(op + 2, v.z * w);
    atomicAdd(op + 3, v.w * w);
  }
}

// ---------------- LeakyReLU + BN ----------------

// stride MUST be a multiple of 128 so each thread owns a fixed column
__global__ void leaky_stats(float* __restrict__ h, float* __restrict__ colsum,
                            float* __restrict__ colsumsq, int n) {
  long total = (long)n * DF;
  long i  = (long)blockIdx.x * blockDim.x + threadIdx.x;
  long st = (long)gridDim.x * blockDim.x;
  int  d  = (int)(i & 127);
  float s = 0.0f, s2 = 0.0f;
  for (; i < total; i += st) {
    float v = h[i];
    v = (v >= 0.0f) ? v : 0.01f * v;
    h[i] = v;
    s  += v;
    s2 += v * v;
  }
  atomicAdd(&colsum[d],   s);
  atomicAdd(&colsumsq[d], s2);
}

__global__ void bn_finalize(const float* __restrict__ colsum, const float* __restrict__ colsumsq,
                            const float* __restrict__ gamma, const float* __restrict__ beta,
                            float* __restrict__ scale, float* __restrict__ shift, float invN) {
  int d = threadIdx.x;
  if (d < DF) {
    float mu  = colsum[d] * invN;
    float var = colsumsq[d] * invN - mu * mu;
    float sc  = gamma[d] * rsqrtf(var + 1e-4f);
    scale[d] = sc;
    shift[d] = beta[d] - mu * sc;
  }
}

// stride MUST be a multiple of 128
__global__ void bn_apply_pool(float* __restrict__ h, const float* __restrict__ scale,
                              const float* __restrict__ shift, const int* __restrict__ batch,
                              float* __restrict__ pool, int layer, int n) {
  long total = (long)n * DF;
  long i  = (long)blockIdx.x * blockDim.x + threadIdx.x;
  long st = (long)gridDim.x * blockDim.x;
  int  d  = (int)(i & 127);
  float sc = scale[d], sh = shift[d];
  int poff = layer * DF + d;
  for (; i < total; i += st) {
    long row = i >> 7;
    float v = h[i] * sc + sh;
    h[i] = v;
    atomicAdd(&pool[(size_t)batch[row] * LDIM + poff], v);
  }
}

__global__ void pool_div(float* __restrict__ pool, const float* __restrict__ counts, int g) {
  int total = g * LDIM;
  int i  = blockIdx.x * blockDim.x + threadIdx.x;
  int st = gridDim.x * blockDim.x;
  for (; i < total; i += st) {
    int gg = i / LDIM;
    pool[i] /= fmaxf(counts[gg], 1.0f);
  }
}

// ---------------- host ----------------

extern "C" void kernel_launch(void* const* d_in, const int* in_sizes, int n_in,
                              void* d_out, int out_size, void* d_ws, size_t ws_size,
                              hipStream_t stream) {
  (void)n_in; (void)ws_size;
  const float* x      = (const float*)d_in[0];
  const int*   ei     = (const int*)  d_in[1];
  const int*   batch  = (const int*)  d_in[2];
  const float* Ws     = (const float*)d_in[3];
  const float* bs     = (const float*)d_in[4];
  const float* gammas = (const float*)d_in[5];
  const float* betas  = (const float*)d_in[6];

  const int N_ = in_sizes[0] / DF;
  const int E_ = in_sizes[1] / 2;
  const long ND = (long)N_ * DF;
  const int G_ = (int)(((long)out_size - ND) / LDIM);

  const int* src = ei;
  const int* dst = ei + E_;

  float* pool  = (float*)d_out;                    // [G, L*D]
  float* hlast = (float*)d_out + (size_t)G_ * LDIM; // [N, D] final layer output

  float* ws    = (float*)d_ws;
  float* buf0  = ws;                 // GEMM output
  float* buf1  = ws + ND;            // aggregated / BN'd features
  float* dinv  = ws + 2 * ND;        // [N]
  float* colsum   = dinv + N_;       // [128]
  float* colsumsq = colsum + DF;     // [128]
  float* scale    = colsumsq + DF;   // [128]
  float* shift    = scale + DF;      // [128]
  float* counts   = shift + DF;      // [G]

  // one-time precompute
  zero_f<<<64, 256, 0, stream>>>(dinv, N_);
  zero_f<<<8, 256, 0, stream>>>(counts, G_);
  zero_f<<<256, 256, 0, stream>>>(pool, (long)G_ * LDIM);
  count_nodes<<<256, 256, 0, stream>>>(batch, counts, N_);
  degree_edges<<<1024, 256, 0, stream>>>(dst, dinv, E_);
  make_dinv<<<256, 256, 0, stream>>>(dinv, N_);

  const int tiles = (N_ + 15) / 16;
  const int gemm_blocks = (tiles + 3) / 4;

  for (int layer = 0; layer < NLAYERS; ++layer) {
    const float* hin  = (layer == 0) ? x : buf1;
    float*       hout = (layer < NLAYERS - 1) ? buf1 : hlast;

    gemm_wmma_f32<<<gemm_blocks, 128, 0, stream>>>(hin, Ws + (size_t)layer * DF * DF, buf0, N_);
    init_self_bias<<<512, 256, 0, stream>>>(buf0, dinv, bs + layer * DF, hout, N_);
    agg_edges<<<2048, 256, 0, stream>>>(buf0, dinv, src, dst, hout, E_);
    zero_f<<<1, 256, 0, stream>>>(colsum, 2 * DF);   // colsum + colsumsq (contiguous)
    leaky_stats<<<512, 256, 0, stream>>>(hout, colsum, colsumsq, N_);
    bn_finalize<<<1, 128, 0, stream>>>(colsum, colsumsq, gammas + layer * DF,
                                       betas + layer * DF, scale, shift, 1.0f / (float)N_);
    bn_apply_pool<<<512, 256, 0, stream>>>(hout, scale, shift, batch, pool, layer, N_);
  }

  pool_div<<<256, 256, 0, stream>>>(pool, counts, G_);
}